// Sampler_17025250362068
// MI455X (gfx1250) — compile-verified
//
#include <hip/hip_runtime.h>
#include <hip/hip_bf16.h>
#include <stdint.h>

// ---------------------------------------------------------------------------
// Problem constants (from reference): P=S=16, H=W=256 -> nh=nw=16, L=256,
// N=128, D=512, kch=ceil(512/256)=2, B=4, C=3.
// ---------------------------------------------------------------------------
#define PB    16      // patch
#define NWB   16      // patches per row
#define LB    256     // patches per batch
#define NSEL  128     // sampled patches
#define BB    4       // batch
#define CCH   3       // x channels
#define KCH   2       // pos channels gathered
#define HW    256
#define XELEM (CCH * PB * PB)   // 768 floats per patch from x
#define PELEM 512               // floats per patch from pos

typedef __attribute__((ext_vector_type(2))) float v2f;
typedef __attribute__((ext_vector_type(8))) float v8f;

// ---------------------------------------------------------------------------
// Kernel A: per-patch mask sum via the matrix pipe + hash-Gumbel score.
// One wave (32 lanes) per (batch, patch-row-stripe).  Computes
//   Q(16x16) = M(16x256) x BlockDiag(256x16)  with v_wmma_f32_16x16x4_f32,
// where column p of Q is the per-row sum of patch p in this stripe.
// C-layout: lane holds column N=lane%16, rows M=v+8*(lane/16)  -> per-patch
// sum = local 8-row sum + shfl_xor(16).
// ---------------------------------------------------------------------------
__global__ __launch_bounds__(32)
void mask_score_kernel(const float* __restrict__ mask, float* __restrict__ scores) {
    const int b    = blockIdx.x >> 4;    // batch
    const int r    = blockIdx.x & 15;    // patch-row stripe
    const int lane = threadIdx.x;        // 0..31
    const int m    = lane & 15;
    const int hi   = lane >> 4;

    const float* mrow = mask + ((size_t)b << 16) + (size_t)(r * PB + m) * HW;

    v8f acc = {};
    #pragma unroll 8
    for (int k = 0; k < 64; ++k) {
        // A (16x4 f32): lanes 0-15 hold K=0,1 ; lanes 16-31 hold K=2,3
        const int c0 = 4 * k + 2 * hi;
        v2f a = *(const v2f*)(mrow + c0);                 // global_load_b64
        // B (4x16): block-diagonal patch selector; chunk k feeds patch k>>2,
        // so every row of this chunk is 1 in column (k>>2) and 0 elsewhere.
        const float sel = (m == (k >> 2)) ? 1.0f : 0.0f;
        v2f bv; bv.x = sel; bv.y = sel;
        acc = __builtin_amdgcn_wmma_f32_16x16x4_f32(
            /*neg_a=*/false, a, /*neg_b=*/false, bv,
            /*c_mod=*/(short)0, acc, /*reuse_a=*/false, /*reuse_b=*/false);
    }

    float partial = acc[0] + acc[1] + acc[2] + acc[3] +
                    acc[4] + acc[5] + acc[6] + acc[7];      // 8 rows of my column
    partial += __shfl_xor(partial, 16, 32);                 // + other 8 rows

    if (lane < 16) {
        const int l     = r * NWB + m;                      // unfold column index
        const bool valid = partial >= (256.0f - 0.5f);      // mean >= TH (=1.0)
        // deterministic hash-Gumbel (stand-in for jax.random.gumbel(key 42))
        unsigned s = (unsigned)(b * LB + l) * 2654435761u + 42u;
        s ^= s >> 16; s *= 0x7feb352du;
        s ^= s >> 15; s *= 0x846ca68bu;
        s ^= s >> 16;
        const float u = ((float)s + 0.5f) * (1.0f / 4294967296.0f);  // (0,1)
        const float g = -__logf(-__logf(u));
        scores[b * LB + l] = valid ? g : -__builtin_inff();
    }
}

// ---------------------------------------------------------------------------
// Kernel B: per-batch exact top-128 by rank counting (total order: score desc,
// index asc), then ascending-index compaction via prefix count.  L=256 so the
// O(L^2) LDS scheme is ~65K compares per batch -- negligible.
// ---------------------------------------------------------------------------
__global__ __launch_bounds__(256)
void topk_kernel(const float* __restrict__ scores, int* __restrict__ idx) {
    __shared__ float         s[LB];
    __shared__ unsigned char sel[LB];
    const int b = blockIdx.x;
    const int i = threadIdx.x;

    const float mine = scores[b * LB + i];
    s[i] = mine;
    __syncthreads();

    int rank = 0;
    #pragma unroll 8
    for (int j = 0; j < LB; ++j) {
        const float o = s[j];
        rank += (o > mine) || (o == mine && j < i);
    }
    const bool selected = rank < NSEL;    // ranks unique -> exactly 128 winners
    sel[i] = selected ? 1 : 0;
    __syncthreads();

    if (selected) {
        int pos = 0;
        for (int j = 0; j < i; ++j) pos += sel[j];
        idx[b * NSEL + pos] = i;          // ascending order by construction
    }
}

// ---------------------------------------------------------------------------
// Kernel C: patch gather on the CDNA5 async data path.
// One 320-thread block per (b, n).  5 channels (3 from x, 2 from pos) x 16
// rows x 4 segments of 16B = 320 async b128 global->LDS transfers; then
// s_wait_asynccnt + barrier; then 320 async b128 LDS->global stores laid out
// exactly in the reference's channel-major flatten order.  s_endpgm's
// implicit wait-idle covers the outstanding async stores.
// ---------------------------------------------------------------------------
__global__ __launch_bounds__(320)
void gather_kernel(const float* __restrict__ x, const float* __restrict__ pos,
                   const int* __restrict__ idx, float* __restrict__ out) {
    __shared__ float smem[(XELEM + PELEM)];   // 1280 floats = 5120 B

    const int bn = blockIdx.x;       // 0..511
    const int b  = bn >> 7;
    const int n  = bn & 127;
    const int t  = threadIdx.x;      // 0..319

    const int l  = idx[b * NSEL + n];
    const int ri = (l >> 4) << 4;    // top row of patch
    const int ci = (l & 15) << 4;    // left col of patch

    const int ch5 = t >> 6;          // 0..4  (0-2: x channels, 3-4: pos channels)
    const int rem = t & 63;
    const int pr  = rem >> 2;        // patch row 0..15
    const int seg = rem & 3;         // 16B segment within the 64B row

    const float* src;
    if (ch5 < CCH) {
        src = x   + (((size_t)(b * CCH + ch5))        << 16) + (size_t)(ri + pr) * HW + ci + seg * 4;
    } else {
        src = pos + (((size_t)(b * 512 + (ch5 - CCH))) << 16) + (size_t)(ri + pr) * HW + ci + seg * 4;
    }

    // LDS slot 4*t is exactly output element index (x part: ch*256+pr*16+4*seg;
    // pos part: 768 + (ch-3)*256+pr*16+4*seg) -- identity layout.
    const unsigned lds = (unsigned)(uintptr_t)&smem[t * 4];

    asm volatile("global_load_async_to_lds_b128 %0, %1, off"
                 :: "v"(lds), "v"((unsigned long long)(uintptr_t)src)
                 : "memory");
    asm volatile("s_wait_asynccnt 0x0" ::: "memory");
    __syncthreads();

    float* dst;
    if (t < (XELEM / 4)) {
        // x_sample[b, n, :] block of d_out
        dst = out + (size_t)(b * NSEL + n) * XELEM + t * 4;
    } else {
        // pos_sample[b, n, :] block, after all of x_sample
        dst = out + (size_t)BB * NSEL * XELEM
                  + (size_t)(b * NSEL + n) * PELEM + (t - XELEM / 4) * 4;
    }
    asm volatile("global_store_async_from_lds_b128 %0, %1, off"
                 :: "v"((unsigned long long)(uintptr_t)dst), "v"(lds)
                 : "memory");
    // implicit s_wait_idle at s_endpgm drains ASYNCcnt for the stores
}

// ---------------------------------------------------------------------------
// Launch: A -> B -> C on the given stream.  Workspace: 1024 f32 scores + 512
// int indices.
// ---------------------------------------------------------------------------
extern "C" void kernel_launch(void* const* d_in, const int* in_sizes, int n_in,
                              void* d_out, int out_size, void* d_ws, size_t ws_size,
                              hipStream_t stream) {
    const float* x    = (const float*)d_in[0];   // [4,3,256,256]
    const float* mask = (const float*)d_in[1];   // [4,1,256,256]
    const float* pos  = (const float*)d_in[2];   // [4,512,256,256]
    float* out = (float*)d_out;

    float* scores = (float*)d_ws;                              // 4*256 f32
    int*   idx    = (int*)((char*)d_ws + BB * LB * sizeof(float)); // 4*128 i32

    mask_score_kernel<<<BB * 16, 32,  0, stream>>>(mask, scores);
    topk_kernel      <<<BB,      256, 0, stream>>>(scores, idx);
    gather_kernel    <<<BB * NSEL, 320, 0, stream>>>(x, pos, idx, out);
}